// MoNet_36687610642610
// MI455X (gfx1250) — compile-verified
//
#include <hip/hip_runtime.h>

#define NFEAT   128
#define NHID    96
#define NCLASS  40
#define NLAYERS 2

typedef __attribute__((ext_vector_type(2))) float v2f;
typedef __attribute__((ext_vector_type(8))) float v8f;

// ---------------------------------------------------------------------------
// WMMA fp32 GEMM:  C[M,NB] = A[M,KD] @ B[KD,NB] (+ bias),  KD % 4 == 0.
// Block = 256 threads (8 waves). B panel staged once into LDS via the CDNA5
// async global->LDS path, stored K-pair interleaved so each 16x16x4 B
// fragment is a single ds_load_b64. Each wave computes a 16-row stripe with
// V_WMMA_F32_16X16X4_F32 across NT = ceil(NB/16) column tiles.
// ---------------------------------------------------------------------------
template <int KD, int NB>
__global__ void __launch_bounds__(256)
wmma_gemm_lds(const float* __restrict__ A, const float* __restrict__ Bg,
              const float* __restrict__ bias, float* __restrict__ C, int M) {
  constexpr int NT = (NB + 15) / 16;
  // interleaved pairs: sB[((k>>1)*NB + n)*2 + (k&1)]
  __shared__ float sB[KD * NB];

  const int tid = threadIdx.x;

  // ---- stage B into LDS (async global->LDS, tracked by ASYNCcnt) --------
  // KD*NB is a multiple of 256 for every instantiation -> uniform trip count.
  for (int idx = tid; idx < KD * NB; idx += 256) {
    const int k = idx / NB;          // NB is compile-time: cheap
    const int n = idx - k * NB;
    float* dst = &sB[((((unsigned)k >> 1) * NB + n) << 1) | (k & 1)];
    const unsigned lds_addr = (unsigned)(uintptr_t)dst;  // low 32b = LDS offset
    const float* src = Bg + idx;
    asm volatile("global_load_async_to_lds_b32 %0, %1, off"
                 :: "v"(lds_addr), "v"(src) : "memory");
  }
#if __has_builtin(__builtin_amdgcn_s_wait_asynccnt)
  __builtin_amdgcn_s_wait_asynccnt(0);
#else
  asm volatile("s_wait_asynccnt 0x0" ::: "memory");
#endif
  __syncthreads();

  // ---- per-wave 16-row stripe -------------------------------------------
  const int wv = tid >> 5;
  int m0 = (blockIdx.x * 8 + wv) * 16;
  if (m0 + 16 > M) m0 = M - 16;      // duplicate identical work; EXEC stays full
  const int lane = tid & 31;
  const int half = lane >> 4;        // 0: lanes 0-15, 1: lanes 16-31
  const int r    = lane & 15;

  v8f acc[NT];
  v8f vzero = {};
#pragma unroll
  for (int t = 0; t < NT; ++t) acc[t] = vzero;

  // A 16x4 f32 frag: lane r -> row m0+r; VGPRs hold K = k0+2h, k0+2h+1
  const float* arow = A + (size_t)(m0 + r) * KD + 2 * half;
  const v2f* sBp = (const v2f*)sB;   // pair view, 8-byte aligned

  for (int k0 = 0; k0 < KD; k0 += 4) {
    const v2f a = *(const v2f*)(arow + k0);      // global_load_b64
    const int kp = (k0 >> 1) + half;             // pair-row index (k0+2h)/2
#pragma unroll
    for (int t = 0; t < NT; ++t) {
      const int n  = t * 16 + r;
      const int nc = (n < NB) ? n : (NB - 1);
      v2f b = sBp[kp * NB + nc];                 // ds_load_b64
      b.x = (n < NB) ? b.x : 0.0f;               // folds away when NB%16==0
      b.y = (n < NB) ? b.y : 0.0f;
      acc[t] = __builtin_amdgcn_wmma_f32_16x16x4_f32(
          false, a, false, b, (short)0, acc[t], false, false);
    }
  }

  // C/D frag: VGPR i, half h -> row m0 + i + 8h, col n
#pragma unroll
  for (int t = 0; t < NT; ++t) {
    const int n = t * 16 + r;
    if (n < NB) {
      const float bv = bias ? bias[n] : 0.0f;
#pragma unroll
      for (int i = 0; i < 8; ++i) {
        const int m = m0 + i + 8 * half;
        C[(size_t)m * NB + n] = acc[t][i] + bv;
      }
    }
  }
}

// ---------------------------------------------------------------------------
__global__ void zero_f(float* __restrict__ p, long long n) {
  long long i = (long long)blockIdx.x * blockDim.x + threadIdx.x;
  const long long stride = (long long)gridDim.x * blockDim.x;
  for (; i < n; i += stride) p[i] = 0.0f;
}

__global__ void deg_acc(const int* __restrict__ row, const float* __restrict__ w,
                        float* __restrict__ deg, int E) {
  const int e = blockIdx.x * blockDim.x + threadIdx.x;
  if (e < E) atomicAdd(&deg[row[e]], w[e]);
}

__global__ void dinv_k(const float* __restrict__ deg, float* __restrict__ dinv, int n) {
  const int i = blockIdx.x * blockDim.x + threadIdx.x;
  if (i < n) {
    const float d = deg[i];
    dinv[i] = (d > 0.0f) ? rsqrtf(d) : 0.0f;   // deg==0 -> inf -> 0 per reference
  }
}

// gauss[e] = exp(-0.5*(tanh(dinv[row]*w0 + dinv[col]*w1 + b) - mu)^2/(eps+sigma^2))
__global__ void edge_gauss(const int* __restrict__ row, const int* __restrict__ col,
                           const float* __restrict__ dinv,
                           const float* __restrict__ Wp, const float* __restrict__ bp,
                           const float* __restrict__ mu, const float* __restrict__ sg,
                           float* __restrict__ gauss, int E) {
  const int e = blockIdx.x * blockDim.x + threadIdx.x;
  if (e >= E) return;
  const float w0 = Wp[0], w1 = Wp[1], b = bp[0];
  const float p = tanhf(dinv[row[e]] * w0 + dinv[col[e]] * w1 + b);
  const float d = p - mu[0];
  const float s = sg[0];
  gauss[e] = expf(-0.5f * d * d / (1e-15f + s * s));
}

// agg[col[e], :] += gauss[e] * xg[row[e], :]
// e is block-uniform -> row/col/gauss scalarize to s_load; 96 lanes do the
// coalesced gather + global_atomic_add_f32 (xg/agg are L2-resident, 19.2 MB).
__global__ void scatter_msg(const int* __restrict__ row, const int* __restrict__ col,
                            const float* __restrict__ gauss,
                            const float* __restrict__ xg, float* __restrict__ agg,
                            int E) {
  const int f = threadIdx.x;               // 0..NHID-1 (96 threads = 3 waves)
  for (int e = blockIdx.x; e < E; e += gridDim.x) {
    const float g = gauss[e];
    const float v = xg[(long long)row[e] * NHID + f];
    atomicAdd(&agg[(long long)col[e] * NHID + f], g * v);
  }
}

// in-place: agg_io = h_in + relu(agg_io + hroot)   (hroot already has bias)
__global__ void combine_k(const float* __restrict__ hin, const float* __restrict__ hroot,
                          float* __restrict__ agg_io, long long n) {
  const long long i = (long long)blockIdx.x * blockDim.x + threadIdx.x;
  if (i < n) {
    const float v = agg_io[i] + hroot[i];
    agg_io[i] = hin[i] + fmaxf(v, 0.0f);
  }
}

// ---------------------------------------------------------------------------
extern "C" void kernel_launch(void* const* d_in, const int* in_sizes, int n_in,
                              void* d_out, int out_size, void* d_ws, size_t ws_size,
                              hipStream_t stream) {
  (void)n_in; (void)out_size; (void)ws_size;

  const float* h0    = (const float*)d_in[0];
  const int*   eidx  = (const int*)  d_in[1];
  const float* ew    = (const float*)d_in[2];
  const float* Wemb  = (const float*)d_in[3];
  const float* bemb  = (const float*)d_in[4];
  const float* Wp    = (const float*)d_in[5];   // [L,2,1]
  const float* bp    = (const float*)d_in[6];   // [L,1]
  const float* Wg    = (const float*)d_in[7];   // [L,96,96]
  const float* mu    = (const float*)d_in[8];   // [L,1,1]
  const float* sg    = (const float*)d_in[9];   // [L,1,1]
  const float* Wroot = (const float*)d_in[10];  // [L,96,96]
  const float* bcv   = (const float*)d_in[11];  // [L,96]
  const float* Wout  = (const float*)d_in[12];  // [96,40]
  const float* bout  = (const float*)d_in[13];  // [40]

  const int E = in_sizes[2];
  const int N = in_sizes[0] / NFEAT;
  const int* row = eidx;          // edge_index[0]
  const int* col = eidx + E;      // edge_index[1]

  // Workspace (floats): buf0, buf1 (ping-pong h / agg), scratch (xg then hroot),
  // deg, dinv, gauss. Total ~61 MB.
  float* p = (float*)d_ws;
  float* buf0 = p;  p += (size_t)N * NHID;
  float* buf1 = p;  p += (size_t)N * NHID;
  float* scr  = p;  p += (size_t)N * NHID;
  float* deg  = p;  p += N;
  float* dinv = p;  p += N;
  float* gas  = p;  p += E;

  const long long NH = (long long)N * NHID;
  const int gemm_blks = (N / 16 + 7) / 8;        // 8 waves/block, 16 rows/wave
  const int eb  = (E + 255) / 256;
  const int nb  = (N + 255) / 256;
  const int nhb = (int)((NH + 255) / 256);

  // ---- degrees -> dinv ---------------------------------------------------
  zero_f<<<nb, 256, 0, stream>>>(deg, N);
  deg_acc<<<eb, 256, 0, stream>>>(row, ew, deg, E);
  dinv_k<<<nb, 256, 0, stream>>>(deg, dinv, N);

  // ---- embedding: h = h0 @ Wemb + bemb -> buf0 ---------------------------
  wmma_gemm_lds<NFEAT, NHID><<<gemm_blks, 256, 0, stream>>>(h0, Wemb, bemb, buf0, N);

  float* hin  = buf0;
  float* hout = buf1;
  for (int l = 0; l < NLAYERS; ++l) {
    // gaussian edge weights
    edge_gauss<<<eb, 256, 0, stream>>>(row, col, dinv, Wp + 2 * l, bp + l,
                                       mu + l, sg + l, gas, E);
    // xg = hin @ Wg[l] -> scr
    wmma_gemm_lds<NHID, NHID><<<gemm_blks, 256, 0, stream>>>(
        hin, Wg + (size_t)l * NHID * NHID, nullptr, scr, N);
    // agg (in hout) = scatter_sum(gauss * xg[row]) at col
    zero_f<<<nhb, 256, 0, stream>>>(hout, NH);
    scatter_msg<<<E, NHID, 0, stream>>>(row, col, gas, scr, hout, E);
    // hroot = hin @ Wroot[l] + b_conv[l] -> scr  (stream order: after scatter)
    wmma_gemm_lds<NHID, NHID><<<gemm_blks, 256, 0, stream>>>(
        hin, Wroot + (size_t)l * NHID * NHID, bcv + l * NHID, scr, N);
    // hout = hin + relu(agg + hroot)
    combine_k<<<nhb, 256, 0, stream>>>(hin, scr, hout, NH);
    float* t = hin; hin = hout; hout = t;
  }

  // ---- readout: out = h @ Wout + bout ------------------------------------
  wmma_gemm_lds<NHID, NCLASS><<<gemm_blks, 256, 0, stream>>>(
      hin, Wout, bout, (float*)d_out, N);
}